// DWT3DInverse_20615843021365
// MI455X (gfx1250) — compile-verified
//
#include <hip/hip_runtime.h>

typedef __attribute__((ext_vector_type(2))) float v2f;
typedef __attribute__((ext_vector_type(8))) float v8f;

// DB4 reconstruction low-pass g0 and high-pass g1[k] = (-1)^k * g0[7-k]
__device__ __constant__ float cG0[8] = {
    0.23037781330885523f,  0.7148465705525415f,  0.6308807679295904f,  -0.02798376941698385f,
   -0.18703481171888114f,  0.030841381835986965f, 0.032883011666982945f, -0.010597401784997278f };
__device__ __constant__ float cG1[8] = {
   -0.010597401784997278f, -0.032883011666982945f, 0.030841381835986965f, 0.18703481171888114f,
   -0.02798376941698385f,  -0.6308807679295904f,   0.7148465705525415f,  -0.23037781330885523f };

// 1-D synthesis filter bank as banded fp32 matmul on the matrix pipe:
//   out[m, col] = sum_i A0[m,i]*lo[i,col] + A1[m,i]*hi[i,col],  M = 2n-6
//   A0[m,i] = g0[u], A1[m,i] = g1[u],  u = m-2i+6 in [0,7]
// One wave per 16(M) x 16(col) tile; K swept in chunks of 4 over the band.
// All edge handling lives in the A fragment (A=0 outside band / i>=n), so the
// B loads are unconditional clamped-address loads: no exec divergence in the
// K-loop at all.
__global__ __launch_bounds__(32)
void sfb1d_wmma(const float* __restrict__ lo, const float* __restrict__ hi,
                float* __restrict__ out,
                int n, int M, int cols,
                long long loOS, long long hiOS, long long outOS,
                int inAS, int inCS, int outAS, int outCS)
{
    const int lane = threadIdx.x & 31;
    const int half = lane >> 4;     // 0: K slots {i0, i0+1}; 1: {i0+2, i0+3}
    const int lid  = lane & 15;
    const int mb   = blockIdx.y * 16;
    const int cb   = blockIdx.x * 16;

    const int  m     = mb + lid;          // A-matrix row (M = lane&15, both halves)
    const int  col   = cb + lid;          // B-matrix column
    const bool colOK = col < cols;
    const int  colC  = colOK ? col : (cols - 1);   // clamped (safe) column

    // per-outer base pointers, pre-offset by the clamped column
    const long long cOff = (long long)colC * inCS;
    const float* loP = lo + (long long)blockIdx.z * loOS + cOff;
    const float* hiP = hi + (long long)blockIdx.z * hiOS + cOff;

    // band limits for this 16-row block: mb-1 <= 2i <= mb+21
    const int imin = mb >> 1;
    int imax = (mb + 21) >> 1;
    if (imax > n - 1) imax = n - 1;

    // Each lane/slot has a nonzero filter tap in exactly one K-step
    // (u drops by 8 per step and u0 >= 2). Precompute tap + target step.
    const int ia0 = imin + 2 * half;
    const int u0x = m - 2 * ia0 + 6;      // >= 2 always
    const int u0y = u0x - 2;              // >= 0 always
    const int sx  = u0x >> 3;             // step where slot .x is in-band
    const int sy  = u0y >> 3;             // step where slot .y is in-band
    const float g0x = cG0[u0x & 7], g1x = cG1[u0x & 7];
    const float g0y = cG0[u0y & 7], g1y = cG1[u0y & 7];

    const int nm1 = n - 1;
    v8f acc = {};

    for (int i0 = imin, s = 0; i0 <= imax; i0 += 4, ++s) {
        const int ia = ia0 + 4 * s;       // K index for .x
        const int ib = ia + 1;            // K index for .y

        // A fragments: tap fires only on its target step and only if i < n
        const bool hx = (s == sx) & (ia < n);
        const bool hy = (s == sy) & (ib < n);
        v2f a0, a1;
        a0.x = hx ? g0x : 0.0f;  a0.y = hy ? g0y : 0.0f;
        a1.x = hx ? g1x : 0.0f;  a1.y = hy ? g1y : 0.0f;

        // B fragments: unconditional loads from clamped addresses
        const int iaC  = ia > nm1 ? nm1 : ia;
        const int ibC  = ib > nm1 ? nm1 : ib;
        const int offA = iaC * inAS;
        const int offB = ibC * inAS;
        v2f bLo, bHi;
        bLo.x = loP[offA];  bLo.y = loP[offB];
        bHi.x = hiP[offA];  bHi.y = hiP[offB];

        acc = __builtin_amdgcn_wmma_f32_16x16x4_f32(false, a0, false, bLo,
                                                    (short)0, acc, false, false);
        acc = __builtin_amdgcn_wmma_f32_16x16x4_f32(false, a1, false, bHi,
                                                    (short)0, acc, false, false);
    }

    // C/D layout: VGPR r -> rows mb+r (lanes 0-15) and mb+r+8 (lanes 16-31).
    const int row0 = mb + 8 * half;
    float* p = out + (long long)blockIdx.z * outOS
                   + (long long)col * outCS
                   + (long long)row0 * outAS;

    // Tile-fits check is wave-uniform (blockIdx/params only) -> scalar branch,
    // so the common case is 8 raw coalesced stores with no exec churn.
    if ((mb + 16 <= M) & (cb + 16 <= cols)) {
#pragma unroll
        for (int r = 0; r < 8; ++r) {
            *p = acc[r];
            p += outAS;
        }
    } else {
        const int rowsLeft = M - row0;    // rows this half may store
#pragma unroll
        for (int r = 0; r < 8; ++r) {
            if ((r < rowsLeft) & colOK)
                *p = acc[r];
            p += outAS;
        }
    }
}

extern "C" void kernel_launch(void* const* d_in, const int* in_sizes, int n_in,
                              void* d_out, int out_size, void* d_ws, size_t ws_size,
                              hipStream_t stream)
{
    const float* yl  = (const float*)d_in[0];  // (2,8,36,36,36)
    const float* yh0 = (const float*)d_in[1];  // (2,8,7,66,66,66)
    const float* yh1 = (const float*)d_in[2];  // (2,8,7,36,36,36)
    float* out = (float*)d_out;                // (2,8,126,126,126)
    float* ws  = (float*)d_ws;

    // Workspace layout (sized for level 2, reused by level 1)
    const size_t slotT  = (size_t)16 * 66 * 66 * 126;   // stage-1 outputs (x4)
    const size_t slotT2 = (size_t)16 * 66 * 126 * 126;  // stage-2 outputs (x2)
    float* tb  = ws;
    float* t2b = ws + 4 * slotT;
    float* llb = ws + 4 * slotT + 2 * slotT2;           // level-1 result (16*66^3)

    auto L = [&](const float* lo, const float* hi, float* o,
                 int n, int M, int cols, int outer,
                 long long loOS, long long hiOS, long long outOS,
                 int inAS, int inCS, int outAS, int outCS) {
        dim3 g((unsigned)((cols + 15) / 16), (unsigned)((M + 15) / 16), (unsigned)outer);
        sfb1d_wmma<<<g, dim3(32), 0, stream>>>(lo, hi, o, n, M, cols,
                                               loOS, hiOS, outOS,
                                               inAS, inCS, outAS, outCS);
    };

    // ---------------- Level 1: 36 -> 66 ----------------
    {
        const int D = 36, M = 66;
        const long long D3 = (long long)D * D * D;  // 46656
        const long long H  = 7 * D3;                // yh1 per-(b,c) stride
        const float*    los[4]   = { yl, yh1 + 1 * D3, yh1 + 3 * D3, yh1 + 5 * D3 };
        const long long losOS[4] = { D3, H, H, H };
        const float*    his[4]   = { yh1, yh1 + 2 * D3, yh1 + 4 * D3, yh1 + 6 * D3 };
        // stage 1: axis -1  (outer=16, n=D, cols=D*D; axis stride 1)
        for (int j = 0; j < 4; ++j)
            L(los[j], his[j], tb + j * slotT, D, M, D * D, 16,
              losOS[j], H, (long long)D * D * M, 1, D, 1, M);
        // stage 2: axis -2  (outer=16*D, n=D, cols=M; axis stride M)
        for (int j = 0; j < 2; ++j)
            L(tb + (2 * j) * slotT, tb + (2 * j + 1) * slotT, t2b + j * slotT2,
              D, M, M, 16 * D, (long long)D * M, (long long)D * M,
              (long long)M * M, M, 1, M, 1);
        // stage 3: axis -3  (outer=16, n=D, cols=M*M; axis stride M*M)
        L(t2b, t2b + slotT2, llb, D, M, M * M, 16,
          (long long)D * M * M, (long long)D * M * M,
          (long long)M * M * M, M * M, 1, M * M, 1);
    }

    // ---------------- Level 2: 66 -> 126 ----------------
    {
        const int D = 66, M = 126;
        const long long D3 = (long long)D * D * D;  // 287496
        const long long H  = 7 * D3;                // yh0 per-(b,c) stride
        const float*    los[4]   = { llb, yh0 + 1 * D3, yh0 + 3 * D3, yh0 + 5 * D3 };
        const long long losOS[4] = { D3, H, H, H };
        const float*    his[4]   = { yh0, yh0 + 2 * D3, yh0 + 4 * D3, yh0 + 6 * D3 };
        for (int j = 0; j < 4; ++j)
            L(los[j], his[j], tb + j * slotT, D, M, D * D, 16,
              losOS[j], H, (long long)D * D * M, 1, D, 1, M);
        for (int j = 0; j < 2; ++j)
            L(tb + (2 * j) * slotT, tb + (2 * j + 1) * slotT, t2b + j * slotT2,
              D, M, M, 16 * D, (long long)D * M, (long long)D * M,
              (long long)M * M, M, 1, M, 1);
        L(t2b, t2b + slotT2, out, D, M, M * M, 16,
          (long long)D * M * M, (long long)D * M * M,
          (long long)M * M * M, M * M, 1, M * M, 1);
    }

    (void)in_sizes; (void)n_in; (void)out_size; (void)ws_size;
}